// HeteroGATLayer_62045097558486
// MI455X (gfx1250) — compile-verified
//
#include <hip/hip_runtime.h>
#include <math.h>

typedef __attribute__((ext_vector_type(2))) float v2f;
typedef __attribute__((ext_vector_type(8))) float v8f;
typedef __attribute__((ext_vector_type(4))) int   v4i;

// Problem sizes (fixed by the reference)
constexpr int B_ = 8, N_ = 4096, E_ = 65536, FIN_ = 128, OUT_ = 128, T_ = 4, H_ = 4, C_ = 32;
constexpr int M_ = B_ * N_;                                   // 32768 flattened rows
constexpr long long SZ_H   = (long long)T_ * M_ * OUT_;       // h[t][row][o]
constexpr long long SZ_AL  = (long long)T_ * M_ * H_;         // per (t,row,head) scalars
constexpr long long OFF_H   = 0;
constexpr long long OFF_ALS = OFF_H + SZ_H;
constexpr long long OFF_ALD = OFF_ALS + SZ_AL;
constexpr long long OFF_M   = OFF_ALD + SZ_AL;
constexpr long long OFF_DEN = OFF_M + SZ_AL;
constexpr long long OFF_OUT = OFF_DEN + SZ_AL;                // M_*OUT_ accumulator

// CDNA5 async global->LDS staging, if the toolchain exposes the builtin.
#if defined(__has_builtin)
#  if __has_builtin(__builtin_amdgcn_global_load_async_to_lds_b128)
#    define USE_ASYNC_LDS 1
#  endif
#endif
#ifndef USE_ASYNC_LDS
#  define USE_ASYNC_LDS 0
#endif

__device__ __forceinline__ float lrelu(float v) { return v > 0.0f ? v : 0.2f * v; }

// Sign-aware float atomic max via integer atomics (works for mixed signs).
__device__ __forceinline__ void atomicMaxFloat(float* addr, float val) {
  if (val >= 0.0f) {
    atomicMax((int*)addr, __float_as_int(val));
  } else {
    atomicMin((unsigned int*)addr, (unsigned int)__float_as_int(val));
  }
}

// ---------------------------------------------------------------------------
// Kernel 1: h[t] = x @ W[t] using V_WMMA_F32_16X16X4_F32 (wave32).
// Block = 256 threads = 8 waves, covering 64 rows x 128 cols.
// Per wave: one 16-col tile; B fragments for all K held in 64 VGPRs;
// A fragments for 4 row sub-tiles read from an async-staged LDS x-tile.
// ---------------------------------------------------------------------------
#define LDS_STRIDE 132   // pad 128-float rows -> stride 4 mod 64 banks (conflict-free)

__global__ __launch_bounds__(256) void gemm_wmma_kernel(const float* __restrict__ x,
                                                        const float* __restrict__ W,
                                                        float* __restrict__ h) {
  __shared__ float sx[64 * LDS_STRIDE];

  const int t    = blockIdx.y;
  const int wave = threadIdx.x >> 5;          // 8 waves -> 8 col tiles
  const int lane = threadIdx.x & 31;
  const int m16  = lane & 15;
  const int hi   = lane >> 4;                 // 0 or 1 -> K pair select
  const int rowbase = blockIdx.x * 64;
  const int colg = wave * 16 + m16;

  // --- Stage x[rowbase .. rowbase+63][0..127] into LDS (32 KB) -------------
#pragma unroll
  for (int i = 0; i < 8; ++i) {
    const int f4  = threadIdx.x + i * 256;    // 0..2047 float4 elements
    const int row = f4 >> 5;                  // 0..63
    const int k4  = (f4 & 31) << 2;           // 0,4,...,124
    const float* gsrc = x + (long long)(rowbase + row) * FIN_ + k4;
    float* ldst = &sx[row * LDS_STRIDE + k4];
#if USE_ASYNC_LDS
    __builtin_amdgcn_global_load_async_to_lds_b128(
        (__attribute__((address_space(1))) v4i*)gsrc,
        (__attribute__((address_space(3))) v4i*)ldst, 0, 0);
#else
    *(float4*)ldst = *(const float4*)gsrc;
#endif
  }

  // --- Preload this wave's B fragments for all K (64 VGPRs) ----------------
  const float* Wt = W + (long long)t * FIN_ * OUT_;
  v2f bfrag[32];
#pragma unroll
  for (int ks = 0; ks < 32; ++ks) {
    const int kp = ks * 4 + hi * 2;
    bfrag[ks].x = Wt[kp * OUT_ + colg];       // B[K=kp  ][col]
    bfrag[ks].y = Wt[(kp + 1) * OUT_ + colg]; // B[K=kp+1][col]
  }

#if USE_ASYNC_LDS
  asm volatile("s_wait_asynccnt 0" ::: "memory");
#endif
  __syncthreads();

  // --- Main WMMA loop: 4 row sub-tiles x 32 K-steps ------------------------
  v8f acc[4] = {};
#pragma unroll
  for (int ks = 0; ks < 32; ++ks) {
    const int kp = ks * 4 + hi * 2;
#pragma unroll
    for (int r = 0; r < 4; ++r) {
      const v2f a = *(const v2f*)&sx[(r * 16 + m16) * LDS_STRIDE + kp];
      acc[r] = __builtin_amdgcn_wmma_f32_16x16x4_f32(
          /*neg_a=*/false, a, /*neg_b=*/false, bfrag[ks],
          /*c_mod=*/(short)0, acc[r], /*reuse_a=*/false, /*reuse_b=*/false);
    }
  }

  // D layout: VGPR j -> row j (lanes 0-15) / row j+8 (lanes 16-31), col = lane&15
#pragma unroll
  for (int r = 0; r < 4; ++r) {
#pragma unroll
    for (int j = 0; j < 8; ++j) {
      const int row = rowbase + r * 16 + hi * 8 + j;
      h[(long long)t * M_ * OUT_ + (long long)row * OUT_ + colg] = acc[r][j];
    }
  }
}

// ---------------------------------------------------------------------------
// Kernel 2: attention logits  al_s/al_d[t][row][head] = <h[t,row,head,:], a_*[t,head,:]>
// ---------------------------------------------------------------------------
__global__ void logits_kernel(const float* __restrict__ h,
                              const float* __restrict__ a_src,
                              const float* __restrict__ a_dst,
                              float* __restrict__ als, float* __restrict__ ald) {
  long long idx = (long long)blockIdx.x * blockDim.x + threadIdx.x;  // over T_*M_
  if (idx >= (long long)T_ * M_) return;
  const int t   = (int)(idx / M_);
  const int row = (int)(idx % M_);
  const float* hr = h + (long long)t * M_ * OUT_ + (long long)row * OUT_;
  const float* as = a_src + t * (H_ * C_);
  const float* ad = a_dst + t * (H_ * C_);
#pragma unroll
  for (int hh = 0; hh < H_; ++hh) {
    float ss = 0.0f, sd = 0.0f;
#pragma unroll 8
    for (int c = 0; c < C_; ++c) {
      const float v = hr[hh * C_ + c];
      ss += v * as[hh * C_ + c];
      sd += v * ad[hh * C_ + c];
    }
    als[idx * H_ + hh] = ss;
    ald[idx * H_ + hh] = sd;
  }
}

// ---------------------------------------------------------------------------
// Kernel 3: initialize max (-1e30), denominator (0), output accumulator (0)
// ---------------------------------------------------------------------------
__global__ void init_kernel(float* __restrict__ mx, float* __restrict__ den,
                            float* __restrict__ oacc) {
  long long idx = (long long)blockIdx.x * blockDim.x + threadIdx.x;
  if (idx < SZ_AL) { mx[idx] = -1e30f; den[idx] = 0.0f; }
  if (idx < (long long)M_ * OUT_) oacc[idx] = 0.0f;
}

// ---------------------------------------------------------------------------
// Kernel 4: segment max over (t, dst, head)
// ---------------------------------------------------------------------------
__global__ void edge_max_kernel(const int* __restrict__ ei, const int* __restrict__ et,
                                const float* __restrict__ als, const float* __restrict__ ald,
                                float* __restrict__ mx) {
  long long idx = (long long)blockIdx.x * blockDim.x + threadIdx.x;  // over B_*E_
  if (idx >= (long long)B_ * E_) return;
  const int b = (int)(idx / E_);
  const int e = (int)(idx % E_);
  const int s = ei[(long long)b * 2 * E_ + e];
  const int d = ei[(long long)b * 2 * E_ + E_ + e];
  const int t = et[idx];
  const long long rs = ((long long)t * M_ + (long long)b * N_ + s) * H_;
  const long long rd = ((long long)t * M_ + (long long)b * N_ + d) * H_;
#pragma unroll
  for (int hh = 0; hh < H_; ++hh) {
    const float v = lrelu(als[rs + hh] + ald[rd + hh]);
    atomicMaxFloat(&mx[rd + hh], v);
  }
}

// ---------------------------------------------------------------------------
// Kernel 5: segment sum of exp(logit - max)
// ---------------------------------------------------------------------------
__global__ void edge_den_kernel(const int* __restrict__ ei, const int* __restrict__ et,
                                const float* __restrict__ als, const float* __restrict__ ald,
                                const float* __restrict__ mx, float* __restrict__ den) {
  long long idx = (long long)blockIdx.x * blockDim.x + threadIdx.x;
  if (idx >= (long long)B_ * E_) return;
  const int b = (int)(idx / E_);
  const int e = (int)(idx % E_);
  const int s = ei[(long long)b * 2 * E_ + e];
  const int d = ei[(long long)b * 2 * E_ + E_ + e];
  const int t = et[idx];
  const long long rs = ((long long)t * M_ + (long long)b * N_ + s) * H_;
  const long long rd = ((long long)t * M_ + (long long)b * N_ + d) * H_;
#pragma unroll
  for (int hh = 0; hh < H_; ++hh) {
    const float v  = lrelu(als[rs + hh] + ald[rd + hh]);
    const float ex = expf(v - mx[rd + hh]);
    unsafeAtomicAdd(&den[rd + hh], ex);
  }
}

// ---------------------------------------------------------------------------
// Kernel 6: scatter messages: out[dst, ch] += alpha * h[t, src, ch]
// one thread per (edge, channel); 128 consecutive threads share one edge.
// ---------------------------------------------------------------------------
__global__ void edge_scatter_kernel(const int* __restrict__ ei, const int* __restrict__ et,
                                    const float* __restrict__ als, const float* __restrict__ ald,
                                    const float* __restrict__ mx, const float* __restrict__ den,
                                    const float* __restrict__ h, float* __restrict__ oacc) {
  const long long gid = (long long)blockIdx.x * blockDim.x + threadIdx.x;  // B_*E_*OUT_
  const int ch = (int)(gid & (OUT_ - 1));
  const long long eidx = gid >> 7;                                          // OUT_ == 128
  if (eidx >= (long long)B_ * E_) return;
  const int b = (int)(eidx / E_);
  const int e = (int)(eidx % E_);
  const int s = ei[(long long)b * 2 * E_ + e];
  const int d = ei[(long long)b * 2 * E_ + E_ + e];
  const int t = et[eidx];
  const int hh = ch >> 5;                                                   // C_ == 32
  const long long rs = ((long long)t * M_ + (long long)b * N_ + s) * H_;
  const long long rd = ((long long)t * M_ + (long long)b * N_ + d) * H_;
  const float v     = lrelu(als[rs + hh] + ald[rd + hh]);
  const float ex    = expf(v - mx[rd + hh]);
  const float alpha = ex / (den[rd + hh] + 1e-16f);
  const float val   = alpha * h[(long long)t * M_ * OUT_ +
                                ((long long)b * N_ + s) * OUT_ + ch];
  unsafeAtomicAdd(&oacc[((long long)b * N_ + d) * OUT_ + ch], val);
}

// ---------------------------------------------------------------------------
// Kernel 7: y = LN(x + out + sum_t bias[t]) * gamma + beta ; one wave per row
// ---------------------------------------------------------------------------
__global__ __launch_bounds__(256) void final_ln_kernel(const float* __restrict__ x,
                                                       const float* __restrict__ oacc,
                                                       const float* __restrict__ bias,
                                                       const float* __restrict__ gamma,
                                                       const float* __restrict__ beta,
                                                       float* __restrict__ y) {
  const int row  = blockIdx.x * 8 + (threadIdx.x >> 5);
  const int lane = threadIdx.x & 31;
  const int f    = lane * 4;
  const float* xr  = x    + (long long)row * OUT_ + f;
  const float* orr = oacc + (long long)row * OUT_ + f;

  float v[4];
  float s = 0.0f, ss = 0.0f;
#pragma unroll
  for (int j = 0; j < 4; ++j) {
    const float bsum = bias[f + j] + bias[OUT_ + f + j] +
                       bias[2 * OUT_ + f + j] + bias[3 * OUT_ + f + j];
    v[j] = xr[j] + orr[j] + bsum;
    s  += v[j];
    ss += v[j] * v[j];
  }
#pragma unroll
  for (int off = 16; off > 0; off >>= 1) {
    s  += __shfl_xor(s, off, 32);
    ss += __shfl_xor(ss, off, 32);
  }
  const float mean = s * (1.0f / 128.0f);
  const float var  = ss * (1.0f / 128.0f) - mean * mean;
  const float rstd = rsqrtf(var + 1e-5f);
  float* yr = y + (long long)row * OUT_ + f;
#pragma unroll
  for (int j = 0; j < 4; ++j) {
    yr[j] = (v[j] - mean) * rstd * gamma[f + j] + beta[f + j];
  }
}

// ---------------------------------------------------------------------------
extern "C" void kernel_launch(void* const* d_in, const int* in_sizes, int n_in,
                              void* d_out, int out_size, void* d_ws, size_t ws_size,
                              hipStream_t stream) {
  (void)in_sizes; (void)n_in; (void)out_size; (void)ws_size;
  const float* x     = (const float*)d_in[0];
  const int*   ei    = (const int*)d_in[1];
  const int*   et    = (const int*)d_in[2];
  const float* W     = (const float*)d_in[3];
  const float* a_src = (const float*)d_in[4];
  const float* a_dst = (const float*)d_in[5];
  const float* bias  = (const float*)d_in[6];
  const float* gamma = (const float*)d_in[7];
  const float* beta  = (const float*)d_in[8];
  float* out = (float*)d_out;

  float* ws   = (float*)d_ws;
  float* h    = ws + OFF_H;
  float* als  = ws + OFF_ALS;
  float* ald  = ws + OFF_ALD;
  float* mx   = ws + OFF_M;
  float* den  = ws + OFF_DEN;
  float* oacc = ws + OFF_OUT;

  // init accumulators (must precede edge passes; stream-serialized)
  init_kernel<<<(M_ * OUT_) / 256, 256, 0, stream>>>(mx, den, oacc);
  // WMMA GEMM for all 4 edge types (async LDS staging + fp32 WMMA)
  gemm_wmma_kernel<<<dim3(M_ / 64, T_), 256, 0, stream>>>(x, W, h);
  // attention logits
  logits_kernel<<<(T_ * M_) / 256, 256, 0, stream>>>(h, a_src, a_dst, als, ald);
  // segment softmax + scatter
  edge_max_kernel<<<(B_ * E_) / 256, 256, 0, stream>>>(ei, et, als, ald, mx);
  edge_den_kernel<<<(B_ * E_) / 256, 256, 0, stream>>>(ei, et, als, ald, mx, den);
  edge_scatter_kernel<<<(int)(((long long)B_ * E_ * OUT_) / 256), 256, 0, stream>>>(
      ei, et, als, ald, mx, den, h, oacc);
  // residual + LayerNorm
  final_ln_kernel<<<M_ / 8, 256, 0, stream>>>(x, oacc, bias, gamma, beta, out);
}